// gRNN_47949014892809
// MI455X (gfx1250) — compile-verified
//
#include <hip/hip_runtime.h>

typedef __attribute__((ext_vector_type(2))) float v2f;
typedef __attribute__((ext_vector_type(8))) float v8f;

#define HID 100   // logical hidden size
#define HP  112   // padded hidden size (7 x 16)
#define NJT 7     // N-dimension 16-wide tiles
#define NKT 25    // K-dimension 4-wide tiles (25*4 = 100, exact)

// ---------------------------------------------------------------------------
// Prep: padded q / Q_b / (W_b+Q_b) vectors, plus W pre-swizzled into WMMA
// B-fragment layout: Bf[(kt*NJT+jt)*32 + lane] = {W[j,k0], W[j,k0+1]} with
// k0 = kt*4 + (lane>>4)*2, j = jt*16 + (lane&15). One global_load_b64 per
// fragment in the level kernel, fully coalesced, L0-resident (44.8 KB).
// ---------------------------------------------------------------------------
__global__ void prep_kernel(const float* __restrict__ Qw,
                            const float* __restrict__ Qb,
                            const float* __restrict__ Wb,
                            const float* __restrict__ Ww,
                            float* __restrict__ qpad,
                            float* __restrict__ Qbpad,
                            float* __restrict__ QWbpad,
                            v2f* __restrict__ Bf) {
  const int tid = threadIdx.x;
  for (int j = tid; j < HP; j += blockDim.x) {
    const bool v = (j < HID);
    const float q  = v ? Qw[j] : 0.0f;   // Q_w is (100,1) -> flat
    const float qb = v ? Qb[j] : 0.0f;
    const float wb = v ? Wb[j] : 0.0f;
    qpad[j]   = q;
    Qbpad[j]  = qb;
    QWbpad[j] = qb + wb;
  }
  for (int e = tid; e < NKT * NJT * 32; e += blockDim.x) {
    const int kt   = e / (NJT * 32);
    const int rem  = e % (NJT * 32);
    const int jt   = rem / 32;
    const int lane = rem % 32;
    const int k0 = kt * 4 + (lane >> 4) * 2;   // always < 100
    const int j  = jt * 16 + (lane & 15);
    v2f b;
    b.x = (j < HID) ? Ww[j * HID + k0]     : 0.0f;
    b.y = (j < HID) ? Ww[j * HID + k0 + 1] : 0.0f;
    Bf[e] = b;
  }
}

// ---------------------------------------------------------------------------
// One tree level: h_out[r,:] = relu((h[2r]+h[2r+1]) @ W^T + W_b + t[r]*q + Q_b)
// 8 waves/block, one 16x112 output tile per wave, fp32 WMMA 16x16x4.
// Dual interleaved accumulator chains (2 j-tiles) for XDL ILP; B fragments
// batched into registers (one wait, then WMMAs run free); A fragments come
// from LDS in-loop (ds_load_2addr_b64, latency hidden by the chain).
// __launch_bounds__(256,1): allow ~2x VGPRs so nothing spills to scratch.
// leafChildren==1: children recomputed as relu(times*q+Q_b) on the fly.
// ---------------------------------------------------------------------------
__global__ void __launch_bounds__(256, 1)
level_kernel(const float* __restrict__ times, int toff,
             const float* __restrict__ hin, int nIn,
             float* __restrict__ hout, int mOut,
             const float* __restrict__ qpad,
             const float* __restrict__ Qbpad,
             const float* __restrict__ QWbpad,
             const v2f* __restrict__ Bf,
             int leafChildren) {
  __shared__ float smem[8 * 16 * HP];  // 57344 B: wave-private child_sum slices

  const int lane = threadIdx.x & 31;
  const int w    = threadIdx.x >> 5;
  const int r0   = (blockIdx.x * 8 + w) * 16;   // output-row base of this wave
  float* S = smem + w * 16 * HP;

  // ---- Stage child_sum rows r0..r0+15 into LDS (float4 granularity) ----
  for (int idx = lane; idx < 16 * (HP / 4); idx += 32) {
    const int row = idx / (HP / 4);
    const int c4  = (idx % (HP / 4)) * 4;
    const int r   = r0 + row;
    int cr0 = 2 * r;     if (cr0 > nIn - 1) cr0 = nIn - 1;
    int cr1 = 2 * r + 1; if (cr1 > nIn - 1) cr1 = nIn - 1;
    float4 s;
    if (leafChildren) {
      const float t0 = times[cr0];
      const float t1 = times[cr1];
      const float4 qv = *(const float4*)(qpad + c4);
      const float4 bv = *(const float4*)(Qbpad + c4);
      s.x = fmaxf(fmaf(t0, qv.x, bv.x), 0.0f) + fmaxf(fmaf(t1, qv.x, bv.x), 0.0f);
      s.y = fmaxf(fmaf(t0, qv.y, bv.y), 0.0f) + fmaxf(fmaf(t1, qv.y, bv.y), 0.0f);
      s.z = fmaxf(fmaf(t0, qv.z, bv.z), 0.0f) + fmaxf(fmaf(t1, qv.z, bv.z), 0.0f);
      s.w = fmaxf(fmaf(t0, qv.w, bv.w), 0.0f) + fmaxf(fmaf(t1, qv.w, bv.w), 0.0f);
    } else {
      const float4 a = *(const float4*)(hin + (size_t)cr0 * HP + c4);
      const float4 b = *(const float4*)(hin + (size_t)cr1 * HP + c4);
      s.x = a.x + b.x; s.y = a.y + b.y; s.z = a.z + b.z; s.w = a.w + b.w;
    }
    *(float4*)(S + row * HP + c4) = s;
  }
  __syncthreads();

  const int hf = lane >> 4;     // lane half: selects K pair (A/B) and M half (C)
  const int jl = lane & 15;
  const float* Arow = S + jl * HP + hf * 2;  // base of this lane's A elements

  // t values for this wave's rows, in C-matrix layout (VGPR v -> M = v + hf*8)
  float tv[8];
#pragma unroll
  for (int v = 0; v < 8; ++v) {
    int rr = r0 + v + hf * 8;
    if (rr > mOut - 1) rr = mOut - 1;
    tv[v] = times[toff + rr];
  }

  // ---- 7 j-tiles processed as 3 pairs + 1, dual WMMA chains per pair ----
#pragma unroll
  for (int jp = 0; jp < 4; ++jp) {
    const int  jt0  = 2 * jp;
    const int  jt1  = 2 * jp + 1;
    const bool has1 = (jt1 < NJT);
    const int  j0   = jt0 * 16 + jl;
    const int  j1   = has1 ? (jt1 * 16 + jl) : j0;

    // Batched coalesced B-fragment loads (clause + single wait).
    v2f b0[NKT], b1[NKT];
#pragma unroll
    for (int kt = 0; kt < NKT; ++kt) {
      b0[kt] = Bf[(kt * NJT + jt0) * 32 + lane];
      if (has1) b1[kt] = Bf[(kt * NJT + jt1) * 32 + lane];
    }

    // Seed accumulators with W_b + Q_b + t*q directly in C layout.
    const float qj0 = qpad[j0], bj0 = QWbpad[j0];
    const float qj1 = qpad[j1], bj1 = QWbpad[j1];
    v8f c0, c1;
#pragma unroll
    for (int v = 0; v < 8; ++v) {
      c0[v] = fmaf(tv[v], qj0, bj0);
      c1[v] = fmaf(tv[v], qj1, bj1);
    }

    // Dual interleaved accumulation chains -> 2-way ILP on the matrix pipe.
    // A fragment re-read from LDS each step (pairs merge to ds_load_2addr_b64).
#pragma unroll
    for (int kt = 0; kt < NKT; ++kt) {
      const v2f a = *(const v2f*)(Arow + kt * 4);
      c0 = __builtin_amdgcn_wmma_f32_16x16x4_f32(
          false, a, false, b0[kt], (short)0, c0, false, false);
      if (has1)
        c1 = __builtin_amdgcn_wmma_f32_16x16x4_f32(
            false, a, false, b1[kt], (short)0, c1, false, false);
    }

    // ReLU + store (coalesced 16-lane halves).
#pragma unroll
    for (int v = 0; v < 8; ++v) {
      const int rr = r0 + v + hf * 8;
      if (rr < mOut) {
        hout[(size_t)rr * HP + j0] = fmaxf(c0[v], 0.0f);
        if (has1) hout[(size_t)rr * HP + j1] = fmaxf(c1[v], 0.0f);
      }
    }
  }
}

// ---------------------------------------------------------------------------
// Final projection: out[p] = h @ P_w[p,:] + P_b[p], p < 5.
// ---------------------------------------------------------------------------
__global__ void proj_kernel(const float* __restrict__ h,
                            const float* __restrict__ Pw,
                            const float* __restrict__ Pb,
                            float* __restrict__ out) {
  const int p = threadIdx.x;
  if (p < 5) {
    float acc = Pb[p];
    for (int k = 0; k < HID; ++k) acc = fmaf(h[k], Pw[p * HID + k], acc);
    out[p] = acc;
  }
}

extern "C" void kernel_launch(void* const* d_in, const int* in_sizes, int n_in,
                              void* d_out, int out_size, void* d_ws, size_t ws_size,
                              hipStream_t stream) {
  (void)n_in; (void)out_size; (void)ws_size;
  const float* times = (const float*)d_in[0];
  const float* Qw    = (const float*)d_in[1];
  const float* Qb    = (const float*)d_in[2];
  const float* Ww    = (const float*)d_in[3];
  const float* Wb    = (const float*)d_in[4];
  const float* Pw    = (const float*)d_in[5];
  const float* Pb    = (const float*)d_in[6];

  const int N = (in_sizes[0] + 1) / 2;   // number of leaves (524288)

  float* ws     = (float*)d_ws;
  float* qpad   = ws;                               // 112
  float* Qbpad  = qpad   + HP;                      // 112
  float* QWbpad = Qbpad  + HP;                      // 112 (total 336, 16-aligned)
  v2f*   Bfrag  = (v2f*)(QWbpad + HP);              // 25*7*32 v2f = 44800 B
  float* buf0   = (float*)(Bfrag + NKT * NJT * 32); // (N/2) * 112 rows
  float* buf1   = buf0 + (size_t)(N / 2) * HP;      // (N/4) * 112 rows

  prep_kernel<<<1, 256, 0, stream>>>(Qw, Qb, Wb, Ww, qpad, Qbpad, QWbpad, Bfrag);

  // Level 1: leaf activations fused into the GEMM (no 210 MB leaf buffer).
  int n = N;
  int offset = N;
  int m = n / 2;
  level_kernel<<<(m + 127) / 128, 256, 0, stream>>>(
      times, offset, nullptr, n, buf0, m, qpad, Qbpad, QWbpad, Bfrag, 1);
  offset += m; n = m;

  float* cur = buf0;
  float* nxt = buf1;
  while (n > 1) {
    m = n / 2;
    level_kernel<<<(m + 127) / 128, 256, 0, stream>>>(
        times, offset, cur, n, nxt, m, qpad, Qbpad, QWbpad, Bfrag, 0);
    offset += m; n = m;
    float* t = cur; cur = nxt; nxt = t;
  }

  proj_kernel<<<1, 32, 0, stream>>>(cur, Pw, Pb, (float*)d_out);
}